// HistogramEqualizer_33535104647825
// MI455X (gfx1250) — compile-verified
//
#include <hip/hip_runtime.h>

// Histogram equalization, 64 images x (3*512*512) fp32.
// Memory-bound. With idx-cache path: 201(read) + 251(read+idx) + 251(idx+out)
// = ~703MB traffic -> ~30us floor at 23.3 TB/s HBM.
// Passes: minmax reduce | LDS histogram (+u8 idx cache) | cdf scan | gather.

#define BINS   256
#define IMGN   (3 * 512 * 512)   // 786432 elements per image
#define N4     (IMGN / 4)        // 196608 float4 per image
#define NIMG   64
#define TPB    256
#define BPI    96                // blocks per image for streaming passes
#define WAVES  (TPB / 32)        // wave32: 8 waves per block

// Native 4xf32 vector for nontemporal builtins (HIP float4 is a struct type
// the builtin rejects; ext_vector_type lowers identically to b128).
typedef float f32x4v __attribute__((ext_vector_type(4)));

__device__ __forceinline__ void store_nt_f4(const float4& v, float4* dst) {
    f32x4v t;
    t.x = v.x; t.y = v.y; t.z = v.z; t.w = v.w;
    __builtin_nontemporal_store(t, (f32x4v*)dst);   // global_store_b128 th:NT
}

// Monotone order-preserving float<->uint key (deterministic integer atomics
// for float min/max).
__device__ __forceinline__ unsigned f2key(float f) {
    unsigned u = __float_as_uint(f);
    return (u & 0x80000000u) ? ~u : (u | 0x80000000u);
}
__device__ __forceinline__ float key2f(unsigned k) {
    return __uint_as_float((k & 0x80000000u) ? (k & 0x7FFFFFFFu) : ~k);
}

// Bin index, matching the reference op order exactly:
// norm = (v - mn) * 255 / (mx - mn + 1e-8); clip(0,255); trunc to int
__device__ __forceinline__ int binOf(float v, float mn, float denom) {
    float t = (v - mn) * 255.0f;
    float n = t / denom;
    n = fminf(fmaxf(n, 0.0f), 255.0f);
    return (int)n;
}

__global__ void he_init(unsigned* __restrict__ minb, unsigned* __restrict__ maxb,
                        unsigned* __restrict__ ghist) {
    int i = blockIdx.x * blockDim.x + threadIdx.x;
    if (i < NIMG) { minb[i] = 0xFFFFFFFFu; maxb[i] = 0u; }
    if (i < NIMG * BINS) ghist[i] = 0u;
}

__global__ void he_minmax(const float4* __restrict__ x,
                          unsigned* __restrict__ minb, unsigned* __restrict__ maxb) {
    const int img = blockIdx.y;
    const float4* p = x + (size_t)img * N4;
    unsigned kmin = 0xFFFFFFFFu, kmax = 0u;

    const int stride = gridDim.x * TPB;
    for (int i = blockIdx.x * TPB + threadIdx.x; i < N4; i += stride) {
        if (i + stride < N4) __builtin_prefetch(p + i + stride, 0, 1); // global_prefetch_b8
        float4 v = p[i];                                               // global_load_b128
        unsigned a = f2key(v.x), b = f2key(v.y), c = f2key(v.z), d = f2key(v.w);
        kmin = min(kmin, min(min(a, b), min(c, d)));
        kmax = max(kmax, max(max(a, b), max(c, d)));
    }
    // wave32 butterfly reduce
    #pragma unroll
    for (int off = 16; off > 0; off >>= 1) {
        kmin = min(kmin, (unsigned)__shfl_xor((int)kmin, off, 32));
        kmax = max(kmax, (unsigned)__shfl_xor((int)kmax, off, 32));
    }
    __shared__ unsigned smin[WAVES], smax[WAVES];
    const int wave = threadIdx.x >> 5, lane = threadIdx.x & 31;
    if (lane == 0) { smin[wave] = kmin; smax[wave] = kmax; }
    __syncthreads();
    if (threadIdx.x == 0) {
        unsigned m0 = smin[0], m1 = smax[0];
        #pragma unroll
        for (int w = 1; w < WAVES; ++w) { m0 = min(m0, smin[w]); m1 = max(m1, smax[w]); }
        atomicMin(&minb[img], m0);   // global_atomic_min_u32 (deterministic)
        atomicMax(&maxb[img], m1);   // global_atomic_max_u32
    }
}

// Histogram pass. If idxout != nullptr, also cache the u8 bin index per pixel
// (packed uchar4 per float4 -> coalesced global_store_b32).
__global__ void he_hist(const float4* __restrict__ x,
                        const unsigned* __restrict__ minb, const unsigned* __restrict__ maxb,
                        unsigned* __restrict__ ghist, uchar4* __restrict__ idxout) {
    const int img = blockIdx.y;
    __shared__ unsigned lh[WAVES][BINS];   // per-wave private histograms (8KB LDS)
    for (int i = threadIdx.x; i < WAVES * BINS; i += TPB) ((unsigned*)lh)[i] = 0u;
    __syncthreads();

    const float mn = key2f(minb[img]);
    const float denom = key2f(maxb[img]) - mn + 1e-8f;
    const int wave = threadIdx.x >> 5;
    const float4* p = x + (size_t)img * N4;
    uchar4* qi = idxout ? idxout + (size_t)img * N4 : nullptr;

    const int stride = gridDim.x * TPB;
    for (int i = blockIdx.x * TPB + threadIdx.x; i < N4; i += stride) {
        if (i + stride < N4) __builtin_prefetch(p + i + stride, 0, 1);
        float4 v = p[i];
        int bx = binOf(v.x, mn, denom), by = binOf(v.y, mn, denom);
        int bz = binOf(v.z, mn, denom), bw = binOf(v.w, mn, denom);
        atomicAdd(&lh[wave][bx], 1u);   // ds_add_u32
        atomicAdd(&lh[wave][by], 1u);
        atomicAdd(&lh[wave][bz], 1u);
        atomicAdd(&lh[wave][bw], 1u);
        if (qi) {
            uchar4 u;
            u.x = (unsigned char)bx; u.y = (unsigned char)by;
            u.z = (unsigned char)bz; u.w = (unsigned char)bw;
            qi[i] = u;                  // global_store_b32 (re-read next pass: RT)
        }
    }
    __syncthreads();
    // reduce wave-private copies, flush once per bin per block
    for (int b = threadIdx.x; b < BINS; b += TPB) {
        unsigned s = 0;
        #pragma unroll
        for (int w = 0; w < WAVES; ++w) s += lh[w][b];
        if (s) atomicAdd(&ghist[img * BINS + b], s);       // global_atomic_add_u32
    }
}

__global__ void he_cdf(const unsigned* __restrict__ ghist, float* __restrict__ cdfn) {
    const int img = blockIdx.x;
    const int t = threadIdx.x;      // 256 threads
    __shared__ float s[BINS];
    // counts sum to 786432 < 2^24: every partial sum is exact in fp32
    s[t] = (float)ghist[img * BINS + t];
    __syncthreads();
    // Hillis-Steele inclusive scan over 256 bins
    for (int off = 1; off < BINS; off <<= 1) {
        float v = (t >= off) ? s[t - off] : 0.0f;
        __syncthreads();
        s[t] += v;
        __syncthreads();
    }
    const float c0 = s[0];                     // cdf.min() (monotone cumsum)
    const float d = s[BINS - 1] - c0 + 1e-8f;  // cdf.max()-cdf.min()+eps
    cdfn[img * BINS + t] = (s[t] - c0) / d;
}

// Fast gather: read cached u8 idx (1B/px) instead of re-reading x (4B/px).
__global__ void he_gather_idx(const uchar4* __restrict__ idx,
                              const float* __restrict__ cdfn, float4* __restrict__ out) {
    const int img = blockIdx.y;
    __shared__ float lc[BINS];
    if (threadIdx.x < BINS) lc[threadIdx.x] = cdfn[img * BINS + threadIdx.x];
    __syncthreads();

    const uchar4* p = idx + (size_t)img * N4;
    float4* q = out + (size_t)img * N4;

    const int stride = gridDim.x * TPB;
    for (int i = blockIdx.x * TPB + threadIdx.x; i < N4; i += stride) {
        if (i + stride < N4) __builtin_prefetch(p + i + stride, 0, 1);
        uchar4 u = p[i];                       // global_load_b32
        float4 r;
        r.x = lc[u.x];                         // ds_load_b32 gather
        r.y = lc[u.y];
        r.z = lc[u.z];
        r.w = lc[u.w];
        store_nt_f4(r, &q[i]);                 // global_store_b128 NT (write-once)
    }
}

// Fallback gather (workspace too small for idx cache): re-read x.
__global__ void he_gather_x(const float4* __restrict__ x,
                            const unsigned* __restrict__ minb, const unsigned* __restrict__ maxb,
                            const float* __restrict__ cdfn, float4* __restrict__ out) {
    const int img = blockIdx.y;
    __shared__ float lc[BINS];
    if (threadIdx.x < BINS) lc[threadIdx.x] = cdfn[img * BINS + threadIdx.x];
    __syncthreads();

    const float mn = key2f(minb[img]);
    const float denom = key2f(maxb[img]) - mn + 1e-8f;
    const float4* p = x + (size_t)img * N4;
    float4* q = out + (size_t)img * N4;

    const int stride = gridDim.x * TPB;
    for (int i = blockIdx.x * TPB + threadIdx.x; i < N4; i += stride) {
        if (i + stride < N4) __builtin_prefetch(p + i + stride, 0, 1);
        float4 v = p[i];
        float4 r;
        r.x = lc[binOf(v.x, mn, denom)];
        r.y = lc[binOf(v.y, mn, denom)];
        r.z = lc[binOf(v.z, mn, denom)];
        r.w = lc[binOf(v.w, mn, denom)];
        store_nt_f4(r, &q[i]);
    }
}

extern "C" void kernel_launch(void* const* d_in, const int* in_sizes, int n_in,
                              void* d_out, int out_size, void* d_ws, size_t ws_size,
                              hipStream_t stream) {
    (void)in_sizes; (void)n_in; (void)out_size;
    const float* x = (const float*)d_in[0];
    float* out = (float*)d_out;

    // workspace layout (all consumed slots fully rewritten every call)
    unsigned* minb  = (unsigned*)d_ws;                  // [64]
    unsigned* maxb  = minb + NIMG;                      // [64]
    unsigned* ghist = maxb + NIMG;                      // [64*256]
    float*    cdfn  = (float*)(ghist + NIMG * BINS);    // [64*256]
    const size_t small_bytes = ((size_t)(2 * NIMG + 2 * NIMG * BINS)) * 4;
    // u8 idx cache: 1B per pixel, 16B-aligned start
    const size_t idx_off = (small_bytes + 255) & ~(size_t)255;
    const size_t idx_bytes = (size_t)NIMG * IMGN;       // ~50.3 MB
    const bool use_idx = ws_size >= idx_off + idx_bytes;
    uchar4* idxbuf = use_idx ? (uchar4*)((char*)d_ws + idx_off) : nullptr;

    he_init<<<(NIMG * BINS + TPB - 1) / TPB, TPB, 0, stream>>>(minb, maxb, ghist);

    dim3 grid(BPI, NIMG);
    he_minmax<<<grid, TPB, 0, stream>>>((const float4*)x, minb, maxb);
    he_hist  <<<grid, TPB, 0, stream>>>((const float4*)x, minb, maxb, ghist, idxbuf);
    he_cdf   <<<NIMG, BINS, 0, stream>>>(ghist, cdfn);
    if (use_idx) {
        he_gather_idx<<<grid, TPB, 0, stream>>>(idxbuf, cdfn, (float4*)out);
    } else {
        he_gather_x<<<grid, TPB, 0, stream>>>((const float4*)x, minb, maxb, cdfn, (float4*)out);
    }
}